// DualGNN_90494960926814
// MI455X (gfx1250) — compile-verified
//
#include <hip/hip_runtime.h>

#define NN 200000
#define EE 600000
#define GG 8192
#define HH 128
#define BN_EPS 1e-5f
#define SCAN_T 1024

typedef __attribute__((ext_vector_type(2))) float v2f;
typedef __attribute__((ext_vector_type(8))) float v8f;

// ---------------- in-degree histogram (int atomics) ----------------
__global__ void indeg_k(const int* __restrict__ dst, int* __restrict__ indeg, int E) {
    int i = blockIdx.x * blockDim.x + threadIdx.x;
    if (i < E) atomicAdd(&indeg[dst[i]], 1);
}

__global__ void dinv_k(const int* __restrict__ indeg, float* __restrict__ dinv, int n) {
    int i = blockIdx.x * blockDim.x + threadIdx.x;
    if (i < n) dinv[i] = rsqrtf((float)indeg[i] + 1.0f);   // +1 self-loop
}

// ---------------- single-block exclusive scan of indeg -> off[0..n] ----------------
__global__ void scan_k(const int* __restrict__ indeg, int* __restrict__ off, int n) {
    __shared__ int sums[SCAN_T];
    int t = threadIdx.x;
    const int CH = (n + SCAN_T - 1) / SCAN_T;
    int begin = t * CH;
    int end = begin + CH; if (end > n) end = n;
    int s = 0;
    for (int i = begin; i < end && i >= 0; ++i) s += indeg[i];
    sums[t] = s;
    __syncthreads();
    // Hillis-Steele inclusive scan
    for (int d = 1; d < SCAN_T; d <<= 1) {
        int v = (t >= d) ? sums[t - d] : 0;
        __syncthreads();
        sums[t] += v;
        __syncthreads();
    }
    int run = (t == 0) ? 0 : sums[t - 1];
    for (int i = begin; i < end && i >= 0; ++i) { off[i] = run; run += indeg[i]; }
    if (t == SCAN_T - 1) off[n] = sums[SCAN_T - 1];     // == E
}

// ---------------- CSR fill: csr_src[pos] = src(e), grouped by dst ----------------
__global__ void fill_k(const int* __restrict__ src, const int* __restrict__ dst,
                       int* __restrict__ cursor, int* __restrict__ csr_src, int E) {
    int e = blockIdx.x * blockDim.x + threadIdx.x;
    if (e >= E) return;
    int pos = atomicAdd(&cursor[dst[e]], 1);
    csr_src[pos] = src[e];
}

// ---------------- layer-0 GEMM (K=7) + self-loop init ----------------
__global__ void gemm0_k(const float* __restrict__ x, const float* __restrict__ W0,
                        const float* __restrict__ dinv, float* __restrict__ h,
                        float* __restrict__ agg, int n) {
    int idx = blockIdx.x * blockDim.x + threadIdx.x;
    if (idx >= n * HH) return;
    int f = idx & (HH - 1);
    int node = idx >> 7;
    const float* xr = x + node * 7;
    float s = 0.0f;
#pragma unroll
    for (int k = 0; k < 7; ++k) s += xr[k] * W0[k * HH + f];
    h[idx] = s;
    float d = dinv[node];
    agg[idx] = d * d * s;
}

// ---------------- gather aggregation: agg[n] += dinv[n]*sum(dinv[s]*h[s]) ----------------
__global__ void gather_agg_k(const int* __restrict__ off, const int* __restrict__ csr_src,
                             const float* __restrict__ dinv, const float* __restrict__ h,
                             float* __restrict__ agg, int n) {
    int idx = blockIdx.x * blockDim.x + threadIdx.x;
    if (idx >= n * 32) return;
    int g = idx & 31;                   // 32 groups of 4 features
    int node = idx >> 5;
    int b = off[node], e = off[node + 1];
    float ax = 0.f, ay = 0.f, az = 0.f, aw = 0.f;
    for (int j = b; j < e; ++j) {
        int s = csr_src[j];
        float c = dinv[s];
        float4 v = *((const float4*)(h + (size_t)s * HH) + g);
        ax += c * v.x; ay += c * v.y; az += c * v.z; aw += c * v.w;
    }
    float dn = dinv[node];
    float* ap = agg + (size_t)node * HH + g * 4;
    ap[0] += dn * ax; ap[1] += dn * ay; ap[2] += dn * az; ap[3] += dn * aw;
}

// ---------------- bias + BN(eval) + ReLU ----------------
__global__ void post1_k(const float* __restrict__ agg, const float* __restrict__ b,
                        const float* __restrict__ gamma, const float* __restrict__ beta,
                        const float* __restrict__ mean, const float* __restrict__ var,
                        float* __restrict__ xout, int n) {
    int idx = blockIdx.x * blockDim.x + threadIdx.x;
    if (idx >= n * HH) return;
    int f = idx & (HH - 1);
    float v = agg[idx] + b[f];
    v = (v - mean[f]) * rsqrtf(var[f] + BN_EPS) * gamma[f] + beta[f];
    xout[idx] = fmaxf(v, 0.0f);
}

// ---------------- pack W (128x128) into K-pair interleaved layout for b64 B-frags ----
// Wp[(p*HH + col)*2 + {0,1}] = W[(2p+{0,1})*HH + col]
__global__ void packW_k(const float* __restrict__ W, float* __restrict__ Wp) {
    int idx = blockIdx.x * blockDim.x + threadIdx.x;
    if (idx >= HH * HH / 2) return;
    int col = idx & (HH - 1);
    int p = idx >> 7;
    Wp[((size_t)p * HH + col) * 2 + 0] = W[(2 * p) * HH + col];
    Wp[((size_t)p * HH + col) * 2 + 1] = W[(2 * p + 1) * HH + col];
}

// ---------------- layer-1 GEMM: h1 = X @ W (128x128), f32 WMMA ----------------
// one wave per 16-row block; 8 col-tiles of 16 in registers; in-place X->H1 safe
__global__ void gemm1_wmma_k(const float* X, const float* __restrict__ Wp,
                             const float* __restrict__ dinv, float* H1,
                             float* __restrict__ agg, int nblk) {
    int wave = blockIdx.x * (blockDim.x >> 5) + (threadIdx.x >> 5);
    if (wave >= nblk) return;                 // uniform per wave -> EXEC all-ones inside
    int lane = threadIdx.x & 31;
    int row  = lane & 15;                     // M for A-frag, N for B/C-frags
    int kh   = lane >> 4;                     // K-half select
    long r0  = (long)wave * 16;

    const float* Ar = X + (r0 + row) * HH + 2 * kh;

    v8f acc[8];
    v8f zero = {};
#pragma unroll
    for (int t = 0; t < 8; ++t) acc[t] = zero;

    for (int k = 0; k < HH; k += 4) {
        v2f a;
        a.x = Ar[k];                          // A[M][k + 2*kh]
        a.y = Ar[k + 1];                      // A[M][k + 2*kh + 1]
        const float* Bp = Wp + (((k >> 1) + kh) * HH + row) * 2;
#pragma unroll
        for (int t = 0; t < 8; ++t) {
            v2f b = *(const v2f*)(Bp + t * 32);   // {W[kk][col], W[kk+1][col]}
            acc[t] = __builtin_amdgcn_wmma_f32_16x16x4_f32(
                false, a, false, b, (short)0, acc[t], false, false);
        }
    }

    float dv[8];
#pragma unroll
    for (int v = 0; v < 8; ++v) {
        float d = dinv[r0 + v + 8 * kh];
        dv[v] = d * d;
    }
#pragma unroll
    for (int t = 0; t < 8; ++t) {
        int col = t * 16 + row;
#pragma unroll
        for (int v = 0; v < 8; ++v) {
            long ro = r0 + v + 8 * kh;        // C layout: VGPR v -> M = v + 8*kh
            float val = acc[t][v];
            H1[ro * HH + col]  = val;
            agg[ro * HH + col] = dv[v] * val;
        }
    }
}

// ---------------- bias + BN + ReLU + mean-pool accumulate ----------------
__global__ void post2pool_k(const float* __restrict__ agg, const float* __restrict__ b,
                            const float* __restrict__ gamma, const float* __restrict__ beta,
                            const float* __restrict__ mean, const float* __restrict__ var,
                            const int* __restrict__ batch, float* __restrict__ pool, int n) {
    int idx = blockIdx.x * blockDim.x + threadIdx.x;
    if (idx >= n * HH) return;
    int f = idx & (HH - 1);
    int node = idx >> 7;
    float v = agg[idx] + b[f];
    v = (v - mean[f]) * rsqrtf(var[f] + BN_EPS) * gamma[f] + beta[f];
    v = fmaxf(v, 0.0f);
    unsafeAtomicAdd(&pool[(size_t)batch[node] * HH + f], v);
}

__global__ void cnt_k(const int* __restrict__ batch, float* __restrict__ cnt, int n) {
    int i = blockIdx.x * blockDim.x + threadIdx.x;
    if (i < n) unsafeAtomicAdd(&cnt[batch[i]], 1.0f);
}

// ---------------- final MLP: concat -> fc1+relu -> fc2 ----------------
__global__ void fc_k(const float* __restrict__ pc, const float* __restrict__ cc,
                     const float* __restrict__ ps, const float* __restrict__ cs,
                     const float* __restrict__ w1, const float* __restrict__ b1,
                     const float* __restrict__ w2, const float* __restrict__ b2,
                     float* __restrict__ out) {
    __shared__ float xc[2 * HH];
    __shared__ float ys[HH];
    int g = blockIdx.x, t = threadIdx.x;
    float ic = 1.0f / fmaxf(cc[g], 1.0f);
    float is = 1.0f / fmaxf(cs[g], 1.0f);
    xc[t]      = pc[(size_t)g * HH + t] * ic;
    xc[HH + t] = ps[(size_t)g * HH + t] * is;
    __syncthreads();
    float s = b1[t];
    for (int k = 0; k < 2 * HH; ++k) s += xc[k] * w1[k * HH + t];
    ys[t] = fmaxf(s, 0.0f);
    __syncthreads();
    if (t < 2) {
        float o = b2[t];
        for (int k = 0; k < HH; ++k) o += ys[k] * w2[k * 2 + t];
        out[(size_t)g * 2 + t] = o;
    }
}

extern "C" void kernel_launch(void* const* d_in, const int* in_sizes, int n_in,
                              void* d_out, int out_size, void* d_ws, size_t ws_size,
                              hipStream_t stream) {
    const float* x_c     = (const float*)d_in[0];
    const int*   ei_c    = (const int*)d_in[1];
    const int*   batch_c = (const int*)d_in[2];
    const float* x_s     = (const float*)d_in[3];
    const int*   ei_s    = (const int*)d_in[4];
    const int*   batch_s = (const int*)d_in[5];
    const float* Wc0 = (const float*)d_in[6],  *bc0 = (const float*)d_in[7];
    const float* Wc1 = (const float*)d_in[8],  *bc1 = (const float*)d_in[9];
    const float* bnc_g = (const float*)d_in[10], *bnc_b = (const float*)d_in[11];
    const float* bnc_m = (const float*)d_in[12], *bnc_v = (const float*)d_in[13];
    const float* Ws0 = (const float*)d_in[14], *bs0 = (const float*)d_in[15];
    const float* Ws1 = (const float*)d_in[16], *bs1 = (const float*)d_in[17];
    const float* bns_g = (const float*)d_in[18], *bns_b = (const float*)d_in[19];
    const float* bns_m = (const float*)d_in[20], *bns_v = (const float*)d_in[21];
    const float* fc1w = (const float*)d_in[22], *fc1b = (const float*)d_in[23];
    const float* fc2w = (const float*)d_in[24], *fc2b = (const float*)d_in[25];
    float* out = (float*)d_out;

    // workspace carve-out (256B aligned)
    char* base = (char*)d_ws;
    size_t off_b = 0;
    auto carve = [&](size_t bytes) -> void* {
        void* r = (void*)(base + off_b);
        off_b += (bytes + 255) & ~(size_t)255;
        return r;
    };
    float* bufA    = (float*)carve((size_t)NN * HH * 4);  // h / x1 / h1 (in-place)
    float* bufB    = (float*)carve((size_t)NN * HH * 4);  // agg
    float* dinv    = (float*)carve((size_t)NN * 4);
    int*   indeg   = (int*)  carve((size_t)NN * 4);
    int*   offA    = (int*)  carve((size_t)(NN + 1) * 4);
    int*   cursor  = (int*)  carve((size_t)NN * 4);
    int*   csr_src = (int*)  carve((size_t)EE * 4);
    float* Wp      = (float*)carve((size_t)HH * HH * 4);
    float* pool_c  = (float*)carve((size_t)GG * HH * 4);
    float* pool_s  = (float*)carve((size_t)GG * HH * 4);
    float* cnt_c   = (float*)carve((size_t)GG * 4);
    float* cnt_s   = (float*)carve((size_t)GG * 4);

    hipMemsetAsync(pool_c, 0, (size_t)GG * HH * 4, stream);
    hipMemsetAsync(pool_s, 0, (size_t)GG * HH * 4, stream);
    hipMemsetAsync(cnt_c, 0, (size_t)GG * 4, stream);
    hipMemsetAsync(cnt_s, 0, (size_t)GG * 4, stream);

    const int TB = 256;
    const int gE   = (EE + TB - 1) / TB;
    const int gN   = (NN + TB - 1) / TB;
    const int gNH  = (NN * HH + TB - 1) / TB;
    const int gN32 = (NN * 32 + TB - 1) / TB;
    const int nblk = NN / 16;                     // 12500 row blocks
    const int gW   = (nblk + 7) / 8;              // 8 waves per 256-thread block

    auto run_branch = [&](const float* x, const int* ei, const int* batch,
                          const float* W0, const float* b0,
                          const float* W1, const float* b1,
                          const float* g_, const float* be_,
                          const float* m_, const float* v_,
                          float* pool, float* cnt) {
        // ---- CSR build (once per branch, reused by both layers) ----
        hipMemsetAsync(indeg, 0, (size_t)NN * 4, stream);
        indeg_k<<<gE, TB, 0, stream>>>(ei + EE, indeg, EE);
        dinv_k<<<gN, TB, 0, stream>>>(indeg, dinv, NN);
        scan_k<<<1, SCAN_T, 0, stream>>>(indeg, offA, NN);
        hipMemcpyAsync(cursor, offA, (size_t)NN * 4, hipMemcpyDeviceToDevice, stream);
        fill_k<<<gE, TB, 0, stream>>>(ei, ei + EE, cursor, csr_src, EE);
        packW_k<<<(HH * HH / 2 + TB - 1) / TB, TB, 0, stream>>>(W1, Wp);
        // ---- layer 0 ----
        gemm0_k<<<gNH, TB, 0, stream>>>(x, W0, dinv, bufA, bufB, NN);
        gather_agg_k<<<gN32, TB, 0, stream>>>(offA, csr_src, dinv, bufA, bufB, NN);
        post1_k<<<gNH, TB, 0, stream>>>(bufB, b0, g_, be_, m_, v_, bufA, NN);
        // ---- layer 1 (WMMA f32) ----
        gemm1_wmma_k<<<gW, TB, 0, stream>>>(bufA, Wp, dinv, bufA, bufB, nblk);
        gather_agg_k<<<gN32, TB, 0, stream>>>(offA, csr_src, dinv, bufA, bufB, NN);
        post2pool_k<<<gNH, TB, 0, stream>>>(bufB, b1, g_ + HH, be_ + HH, m_ + HH, v_ + HH,
                                            batch, pool, NN);
        cnt_k<<<gN, TB, 0, stream>>>(batch, cnt, NN);
    };

    run_branch(x_c, ei_c, batch_c, Wc0, bc0, Wc1, bc1, bnc_g, bnc_b, bnc_m, bnc_v, pool_c, cnt_c);
    run_branch(x_s, ei_s, batch_s, Ws0, bs0, Ws1, bs1, bns_g, bns_b, bns_m, bns_v, pool_s, cnt_s);

    fc_k<<<GG, HH, 0, stream>>>(pool_c, cnt_c, pool_s, cnt_s, fc1w, fc1b, fc2w, fc2b, out);
    (void)in_sizes; (void)n_in; (void)out_size; (void)ws_size;
}